// VQVAE_62921270887009
// MI455X (gfx1250) — compile-verified
//
#include <hip/hip_runtime.h>
#include <hip/hip_bf16.h>
#include <math.h>

typedef __attribute__((ext_vector_type(2))) float v2f;
typedef __attribute__((ext_vector_type(8))) float v8f;

#define N_TOK   65536
#define HID     256
#define D_EMB   128
#define K_CODES 2048
#define A_DIM   6

// LDS row strides (dwords) chosen for bank-conflict-free WMMA fragment reads.
#define EPAD 132   // dist: [code][k], bank = (4*code + k) % 64
#define WPAD 72    // gemm: [k][col],  bank = (8*k + col) % 64

// Low 32 bits of a flat shared-memory pointer are the wave-relative LDS byte
// offset on CDNA5 (flat aperture mapping).
__device__ __forceinline__ unsigned lds_off(const void* p) {
    return (unsigned)(unsigned long long)p;
}

// Async DMA of 16 contiguous bytes global -> LDS (ASYNCcnt-tracked, no VGPR data).
__device__ __forceinline__ void async_copy_b128(unsigned ldsByteOff, const void* gaddr) {
    unsigned long long ga = (unsigned long long)gaddr;
    asm volatile("global_load_async_to_lds_b128 %0, %1, off"
                 :: "v"(ldsByteOff), "v"(ga) : "memory");
}
__device__ __forceinline__ void async_wait0() {
    asm volatile("s_wait_asynccnt 0x0" ::: "memory");
}

// ---------------------------------------------------------------------------
// Encoder layer 1: [N,6] @ [6,256] + b, ReLU.  K=6: VALU is the right tool.
// ---------------------------------------------------------------------------
__global__ void enc1_kernel(const float* __restrict__ action,
                            const float* __restrict__ W,
                            const float* __restrict__ b,
                            float* __restrict__ out) {
    int gid = blockIdx.x * blockDim.x + threadIdx.x;
    int n = gid >> 8;
    int j = gid & 255;
    float s = b[j];
#pragma unroll
    for (int k = 0; k < A_DIM; ++k)
        s = fmaf(action[n * A_DIM + k], W[k * HID + j], s);
    out[(size_t)n * HID + j] = fmaxf(s, 0.0f);
}

// ---------------------------------------------------------------------------
// fp32 WMMA GEMM with async-staged LDS B panel.
// Block = 8 waves: 128 rows x 64 cols. B panel (W[k0..k0+127][col0..col0+15])
// is DMA'd into LDS once per 128-K chunk and shared by all 8 waves (8x L2
// traffic cut vs per-wave fetch). Fragment maps:
//   A (per lane): row = lane%16, K = 2*(lane/16)+{0,1}  (float2 global load)
//   B (per lane): col = lane%16, K = 2*(lane/16)+{0,1}  (two ds_load_b32)
//   C/D: VGPR r <-> row r + 8*(lane/16), col lane%16
// ---------------------------------------------------------------------------
__global__ __launch_bounds__(256)
void gemm_bias_act_kernel(const float* __restrict__ in,
                          const float* __restrict__ W,
                          const float* __restrict__ bias,
                          float* __restrict__ out,
                          int K, int N, int relu) {
    __shared__ float ldsW[128 * WPAD];          // 36 KB
    const int tid  = threadIdx.x;
    const int lane = tid & 31;
    const int wave = tid >> 5;
    const int l16  = lane & 15;
    const int half = (lane >> 4) & 1;
    const int tok0 = blockIdx.x * 128 + wave * 16;
    const int col0 = blockIdx.y * 64;

    v8f acc[4];
#pragma unroll
    for (int jj = 0; jj < 4; ++jj)
#pragma unroll
        for (int r = 0; r < 8; ++r) acc[jj][r] = 0.0f;

    const float2* ap = reinterpret_cast<const float2*>(in + (size_t)(tok0 + l16) * K);
    const unsigned ldsBase = lds_off(&ldsW[0]);

    for (int k0 = 0; k0 < K; k0 += 128) {
        __syncthreads();                         // previous chunk fully consumed
        // Stage 128 x 16 panel: 512 x 16B chunks, 2 per thread.
#pragma unroll
        for (int i = 0; i < 2; ++i) {
            int p  = tid + i * 256;              // 0..511
            int kr = p >> 2;                     // panel row 0..127
            int q  = p & 3;                      // 16B quad within row
            async_copy_b128(ldsBase + (unsigned)((kr * WPAD + q * 4) * 4),
                            W + (size_t)(k0 + kr) * N + col0 + q * 4);
        }
        async_wait0();
        __syncthreads();

        for (int kk = 0; kk < 128; kk += 4) {
            float2 av = ap[((k0 + kk) >> 1) + half];
            v2f a; a.x = av.x; a.y = av.y;
#pragma unroll
            for (int jj = 0; jj < 4; ++jj) {
                int c = jj * 16 + l16;
                v2f bfrag;
                bfrag.x = ldsW[(kk + 2 * half + 0) * WPAD + c];
                bfrag.y = ldsW[(kk + 2 * half + 1) * WPAD + c];
                acc[jj] = __builtin_amdgcn_wmma_f32_16x16x4_f32(
                    false, a, false, bfrag, (short)0, acc[jj], false, false);
            }
        }
    }

#pragma unroll
    for (int jj = 0; jj < 4; ++jj) {
        int col = col0 + jj * 16 + l16;
        float bb = bias[col];
#pragma unroll
        for (int r = 0; r < 8; ++r) {
            float v = acc[jj][r] + bb;
            if (relu) v = fmaxf(v, 0.0f);
            out[(size_t)(tok0 + half * 8 + r) * N + col] = v;
        }
    }
}

// ---------------------------------------------------------------------------
// Codebook -0.5*||e_k||^2 (seeds the distance-WMMA accumulator).
// ---------------------------------------------------------------------------
__global__ void norms_kernel(const float* __restrict__ emb, float* __restrict__ nh) {
    int k = blockIdx.x * blockDim.x + threadIdx.x;
    if (k >= K_CODES) return;
    float s = 0.0f;
    for (int d = 0; d < D_EMB; ++d) {
        float v = emb[(size_t)k * D_EMB + d];
        s = fmaf(v, v, s);
    }
    nh[k] = -0.5f * s;
}

// ---------------------------------------------------------------------------
// Distance + argmin (dominant 34.4-GFLOP GEMM).
// Block = 8 waves = 128 tokens; enc tile per wave preloaded in 64 VGPRs.
// Each 64-code chunk of emb (32 KB) is async-DMA'd into LDS once per block
// (8x traffic cut), then each wave runs 128 WMMAs against it. Accumulator is
// seeded with -0.5||e||^2 so argmax(acc) == argmin(distance); ties resolve
// to the smaller index (monotone traversal + idx tie-break), like jnp.argmin.
// ---------------------------------------------------------------------------
__global__ __launch_bounds__(256)
void dist_argmin_kernel(const float* __restrict__ enc,
                        const float* __restrict__ emb,
                        const float* __restrict__ negHalfNorm,
                        int* __restrict__ idxOut) {
    __shared__ float ldsE[64 * EPAD];           // 33 KB
    const int tid  = threadIdx.x;
    const int lane = tid & 31;
    const int wave = tid >> 5;
    const int l16  = lane & 15;
    const int half = (lane >> 4) & 1;
    const int tok0 = blockIdx.x * 128 + wave * 16;

    // Preload A fragments for all 32 K-steps (D_EMB/4).
    v2f a_reg[32];
    const float2* encp = reinterpret_cast<const float2*>(enc + (size_t)(tok0 + l16) * D_EMB);
#pragma unroll
    for (int t = 0; t < 32; ++t) {
        float2 v = encp[2 * t + half];
        a_reg[t].x = v.x; a_reg[t].y = v.y;
    }

    float best[8];
    int   bidx[8];
#pragma unroll
    for (int r = 0; r < 8; ++r) { best[r] = -3.4e38f; bidx[r] = 0; }

    const int c_st = tid & 63;                  // staged code within chunk
    const int qb   = tid >> 6;                  // 16B quad phase (0..3)
    const unsigned ldsBase = lds_off(&ldsE[0]);

    for (int cb = 0; cb < K_CODES; cb += 64) {
        // Stage emb[cb..cb+63][0..127]: 2048 x 16B chunks, 8 per thread.
#pragma unroll
        for (int i = 0; i < 8; ++i) {
            int q = qb + 4 * i;                 // dword quad 0..31
            async_copy_b128(ldsBase + (unsigned)((c_st * EPAD + q * 4) * 4),
                            emb + (size_t)(cb + c_st) * D_EMB + q * 4);
        }
        async_wait0();
        __syncthreads();

#pragma unroll
        for (int jj = 0; jj < 4; ++jj) {
            int cl   = jj * 16 + l16;
            int code = cb + cl;
            float nh = negHalfNorm[code];
            v8f acc;
#pragma unroll
            for (int r = 0; r < 8; ++r) acc[r] = nh;

            const float2* bp = reinterpret_cast<const float2*>(&ldsE[cl * EPAD]);
#pragma unroll
            for (int t = 0; t < 32; ++t) {
                float2 bv = bp[2 * t + half];   // one ds_load_b64, conflict-free
                v2f bfrag; bfrag.x = bv.x; bfrag.y = bv.y;
                acc = __builtin_amdgcn_wmma_f32_16x16x4_f32(
                    false, a_reg[t], false, bfrag, (short)0, acc, false, false);
            }
#pragma unroll
            for (int r = 0; r < 8; ++r) {
                if (acc[r] > best[r]) { best[r] = acc[r]; bidx[r] = code; }
            }
        }
        __syncthreads();                        // all reads done before restage
    }

    // Cross-lane (16-lane half-wave) argmax reduction per row.
#pragma unroll
    for (int r = 0; r < 8; ++r) {
        float v = best[r];
        int   i = bidx[r];
        for (int off = 8; off; off >>= 1) {
            float ov = __shfl_xor(v, off, 32);
            int   oi = __shfl_xor(i, off, 32);
            if (ov > v || (ov == v && oi < i)) { v = ov; i = oi; }
        }
        if (l16 == 0) idxOut[tok0 + half * 8 + r] = i;
    }
}

// ---------------------------------------------------------------------------
// Gather q = emb[idx] and per-block partial sums of (q-enc)^2 (2 tokens/block).
// ---------------------------------------------------------------------------
__global__ void gather_vq_kernel(const int* __restrict__ idx,
                                 const float* __restrict__ emb,
                                 const float* __restrict__ enc,
                                 float* __restrict__ q,
                                 float* __restrict__ partial) {
    __shared__ float sdata[256];
    int tid   = threadIdx.x;
    int token = blockIdx.x * 2 + (tid >> 7);
    int d     = tid & 127;
    int code  = idx[token];
    size_t off = (size_t)token * D_EMB + d;
    float qv   = emb[(size_t)code * D_EMB + d];
    q[off] = qv;
    float diff = qv - enc[off];
    sdata[tid] = diff * diff;
    __syncthreads();
    for (int s = 128; s; s >>= 1) {
        if (tid < s) sdata[tid] += sdata[tid + s];
        __syncthreads();
    }
    if (tid == 0) partial[blockIdx.x] = sdata[0];
}

// ---------------------------------------------------------------------------
// Decoder layer 3 (N=6) + tanh + per-block partial sums of (recons-action)^2.
// ---------------------------------------------------------------------------
__global__ void dec3_loss_kernel(const float* __restrict__ d2,
                                 const float* __restrict__ W,
                                 const float* __restrict__ b,
                                 const float* __restrict__ action,
                                 float* __restrict__ partial) {
    __shared__ float sdata[256];
    int tid = threadIdx.x;
    int gid = blockIdx.x * 256 + tid;
    int n = gid / A_DIM;
    int a = gid % A_DIM;
    const float* row = d2 + (size_t)n * HID;
    float s = b[a];
    for (int k = 0; k < HID; ++k)
        s = fmaf(row[k], W[k * A_DIM + a], s);
    float r = tanhf(s);
    float diff = r - action[(size_t)n * A_DIM + a];
    sdata[tid] = diff * diff;
    __syncthreads();
    for (int st = 128; st; st >>= 1) {
        if (tid < st) sdata[tid] += sdata[tid + st];
        __syncthreads();
    }
    if (tid == 0) partial[blockIdx.x] = sdata[0];
}

// ---------------------------------------------------------------------------
// Deterministic single-block finalize: vq = (1+beta)*mean((q-enc)^2) since
// stop_gradient is a no-op in the forward value.
// ---------------------------------------------------------------------------
__global__ void finalize_kernel(const float* __restrict__ pQE, int nQE,
                                const float* __restrict__ pRec, int nRec,
                                float* __restrict__ out) {
    __shared__ float sd[256];
    int tid = threadIdx.x;
    float s = 0.0f;
    for (int i = tid; i < nQE; i += 256) s += pQE[i];
    sd[tid] = s; __syncthreads();
    for (int st = 128; st; st >>= 1) { if (tid < st) sd[tid] += sd[tid + st]; __syncthreads(); }
    float sumQE = sd[0];
    __syncthreads();
    s = 0.0f;
    for (int i = tid; i < nRec; i += 256) s += pRec[i];
    sd[tid] = s; __syncthreads();
    for (int st = 128; st; st >>= 1) { if (tid < st) sd[tid] += sd[tid + st]; __syncthreads(); }
    if (tid == 0) {
        float sumRec = sd[0];
        float m   = sumQE / ((float)N_TOK * (float)D_EMB);
        float vq  = 1.25f * m;
        float rec = sumRec / ((float)N_TOK * (float)A_DIM);
        out[0] = rec + vq;
        out[1] = rec;
        out[2] = vq;
    }
}

// ---------------------------------------------------------------------------
extern "C" void kernel_launch(void* const* d_in, const int* in_sizes, int n_in,
                              void* d_out, int out_size, void* d_ws, size_t ws_size,
                              hipStream_t stream) {
    (void)in_sizes; (void)n_in; (void)out_size; (void)ws_size;

    const float* action = (const float*)d_in[0];
    const float* W_e1 = (const float*)d_in[1];  const float* b_e1 = (const float*)d_in[2];
    const float* W_e2 = (const float*)d_in[3];  const float* b_e2 = (const float*)d_in[4];
    const float* W_e3 = (const float*)d_in[5];  const float* b_e3 = (const float*)d_in[6];
    const float* emb  = (const float*)d_in[7];
    const float* W_d1 = (const float*)d_in[8];  const float* b_d1 = (const float*)d_in[9];
    const float* W_d2 = (const float*)d_in[10]; const float* b_d2 = (const float*)d_in[11];
    const float* W_d3 = (const float*)d_in[12]; const float* b_d3 = (const float*)d_in[13];
    float* out = (float*)d_out;

    char* ws = (char*)d_ws;
    size_t off = 0;
    auto carve = [&](size_t nbytes) { char* p = ws + off; off += (nbytes + 255) & ~(size_t)255; return p; };
    float* H1   = (float*)carve((size_t)N_TOK * HID   * 4);   // h1, later d1
    float* H2   = (float*)carve((size_t)N_TOK * HID   * 4);   // h2, later d2
    float* ENC  = (float*)carve((size_t)N_TOK * D_EMB * 4);
    float* Q    = (float*)carve((size_t)N_TOK * D_EMB * 4);
    float* NH   = (float*)carve((size_t)K_CODES * 4);
    int*   IDX  = (int*)  carve((size_t)N_TOK * 4);
    float* PQE  = (float*)carve((size_t)(N_TOK / 2) * 4);
    float* PREC = (float*)carve((size_t)(N_TOK * A_DIM / 256) * 4);
    float* D1 = H1;
    float* D2 = H2;

    const int mBlk = N_TOK / 128;                 // 512 row-blocks (8 waves each)

    // Encoder
    enc1_kernel<<<N_TOK, 256, 0, stream>>>(action, W_e1, b_e1, H1);
    gemm_bias_act_kernel<<<dim3(mBlk, HID / 64), 256, 0, stream>>>(H1, W_e2, b_e2, H2, HID, HID, 1);
    gemm_bias_act_kernel<<<dim3(mBlk, D_EMB / 64), 256, 0, stream>>>(H2, W_e3, b_e3, ENC, HID, D_EMB, 0);

    // Vector quantization
    norms_kernel<<<K_CODES / 256, 256, 0, stream>>>(emb, NH);
    dist_argmin_kernel<<<mBlk, 256, 0, stream>>>(ENC, emb, NH, IDX);
    gather_vq_kernel<<<N_TOK / 2, 256, 0, stream>>>(IDX, emb, ENC, Q, PQE);

    // Decoder (q_st == q in the forward value)
    gemm_bias_act_kernel<<<dim3(mBlk, HID / 64), 256, 0, stream>>>(Q,  W_d1, b_d1, D1, D_EMB, HID, 1);
    gemm_bias_act_kernel<<<dim3(mBlk, HID / 64), 256, 0, stream>>>(D1, W_d2, b_d2, D2, HID,  HID, 1);
    dec3_loss_kernel<<<N_TOK * A_DIM / 256, 256, 0, stream>>>(D2, W_d3, b_d3, action, PREC);

    finalize_kernel<<<1, 256, 0, stream>>>(PQE, N_TOK / 2, PREC, N_TOK * A_DIM / 256, out);
}